// PatchDominantGradientOrientation_15144054685983
// MI455X (gfx1250) — compile-verified
//
#include <hip/hip_runtime.h>
#include <math.h>

namespace {
constexpr int   kPatch = 32;
constexpr int   kNPix  = 1024;
constexpr int   kBins  = 36;
constexpr float kEps   = 1e-8f;
constexpr float kPi    = 3.14159265358979323846f;

// atan(a)/(2*pi) for a in [0,1]; degree-9 minimax (~1e-5 rad abs error),
// coefficients pre-divided by 2*pi so the whole angle pipeline runs in "turns".
__device__ __forceinline__ float atan_poly_turns(float a) {
    const float C0 =  0.15913360f;
    const float C1 = -0.05256917f;
    const float C2 =  0.02867073f;
    const float C3 = -0.01354927f;
    const float C4 =  0.00331610f;
    float s = a * a;
    float p = __builtin_fmaf(s, C4, C3);
    p = __builtin_fmaf(s, p, C2);
    p = __builtin_fmaf(s, p, C1);
    p = __builtin_fmaf(s, p, C0);
    return a * p;
}
} // namespace

__global__ __launch_bounds__(256, 4)
void patch_dominant_orientation(const float* __restrict__ patches,
                                float* __restrict__ out)
{
    __shared__ __align__(16) float tile[kNPix];      // 32x32 patch, row-major
    __shared__ __align__(16) float hist[kBins * 32]; // lane-privatized, conflict-free
    __shared__ float bins[kBins];
    __shared__ float smarr[kBins];

    const int t    = threadIdx.x;        // 0..255 (8 wave32)
    const int lane = t & 31;

    // ---- CDNA5 async DMA: HBM -> LDS, 16B/lane (4KB tile), ASYNCcnt-tracked.
    //      NT hint: the 256MB input is single-use streaming data.
    {
        const unsigned long long gbase =
            (unsigned long long)patches + (unsigned long long)blockIdx.x * (kNPix * 4ull);
        unsigned lds_off = (unsigned)(unsigned long long)&tile[t * 4];
        unsigned goff    = (unsigned)(t * 16);
        asm volatile("global_load_async_to_lds_b128 %0, %1, %2 th:TH_LOAD_NT"
                     :
                     : "v"(lds_off), "v"(goff), "s"(gbase)
                     : "memory");
    }

    // zero privatized histogram while the DMA is in flight (branch-light b128 stores)
    {
        const float4 z4 = make_float4(0.0f, 0.0f, 0.0f, 0.0f);
        float4* h4 = reinterpret_cast<float4*>(hist);   // 288 float4 slots
        h4[t] = z4;
        if (t < 32) h4[256 + t] = z4;
    }

    asm volatile("s_wait_asynccnt 0" ::: "memory");
    __syncthreads();

    // ---- gradients + soft orientation histogram ----
    // Each thread owns a 4-wide strip of one row: row y, cols [x0, x0+3].
    const int y  = t >> 3;
    const int x0 = (t & 7) << 2;
    const int ym = (y == 0)  ? 0  : y - 1;   // edge-clamped rows (mode='edge' pad)
    const int yp = (y == 31) ? 31 : y + 1;
    const int xl = (x0 == 0)  ? 0  : x0 - 1; // edge-clamped boundary cols
    const int xr = (x0 == 28) ? 31 : x0 + 4;

    const float* rT = &tile[ym * kPatch];
    const float* rM = &tile[y  * kPatch];
    const float* rB = &tile[yp * kPatch];

    float4 T4 = *(const float4*)(rT + x0);   // aligned ds_load_b128, conflict-free
    float4 M4 = *(const float4*)(rM + x0);
    float4 B4 = *(const float4*)(rB + x0);
    float Ta[6] = { rT[xl], T4.x, T4.y, T4.z, T4.w, rT[xr] };
    float Ma[6] = { rM[xl], M4.x, M4.y, M4.z, M4.w, rM[xr] };
    float Ba[6] = { rB[xl], B4.x, B4.y, B4.z, B4.w, rB[xr] };

#pragma unroll
    for (int i = 1; i <= 4; ++i) {
        // Sobel/8 (cross-correlation, matching lax.conv):
        float gx = (Ta[i+1] - Ta[i-1] + 2.0f*(Ma[i+1] - Ma[i-1]) + Ba[i+1] - Ba[i-1]) * 0.125f;
        float gy = (Ba[i-1] - Ta[i-1] + 2.0f*(Ba[i]   - Ta[i])   + Ba[i+1] - Ta[i+1]) * 0.125f;

        float gxe = gx + kEps;
        float mag = __builtin_amdgcn_sqrtf(
            __builtin_fmaf(gy, gy, __builtin_fmaf(gx, gx, kEps)));

        // atan2(gy, gxe) in turn units, (-0.5, 0.5]
        float ax = __builtin_fabsf(gxe), ay = __builtin_fabsf(gy);
        float mn = __builtin_fminf(ax, ay), mx = __builtin_fmaxf(ax, ay);
        float r  = atan_poly_turns(mn * __builtin_amdgcn_rcpf(mx));
        r = (ay  > ax)   ? 0.25f - r : r;
        r = (gxe < 0.0f) ? 0.5f  - r : r;
        r = (gy  < 0.0f) ? -r        : r;

        // o == 36*(atan2 + 1.5*pi)/(2*pi), range [9,45]
        float o   = __builtin_fmaf(r, 36.0f, 27.0f);
        float bo0 = __builtin_floorf(o);
        float wo1 = o - bo0;
        int b0 = (int)bo0;  b0 = (b0 >= kBins) ? b0 - kBins : b0;
        int b1 = b0 + 1;    b1 = (b1 == kBins) ? 0 : b1;

        float w1 = wo1 * mag;
        float w0 = mag - w1;

        // banks (32*bin + lane) mod 64 are distinct across lanes for ANY bin mix
        atomicAdd(&hist[b0 * 32 + lane], w0);   // ds_add_f32, conflict-free
        atomicAdd(&hist[b1 * 32 + lane], w1);
    }

    __syncthreads();

    // ---- reduce lane-private histograms; per-lane rotation -> conflict-free ----
    // (positive global scale: 1/8 Sobel, 1/1024 norm dropped -> argmax invariant)
    if (t < kBins) {
        float s = 0.0f;
#pragma unroll
        for (int k = 0; k < 32; ++k)
            s += hist[t * 32 + ((k + t) & 31)];
        bins[t] = s;
    }
    __syncthreads();

    // ---- circular smoothing in parallel ----
    if (t < kBins) {
        int im = (t == 0)         ? kBins - 1 : t - 1;
        int ip = (t == kBins - 1) ? 0         : t + 1;
        smarr[t] = 0.33f * bins[im] + 0.34f * bins[t] + 0.33f * bins[ip];
    }
    __syncthreads();

    // ---- first-max argmax (matches jnp.argmax tie-break) -> angle ----
    if (t == 0) {
        float best = -INFINITY; int bi = 0;
#pragma unroll 1
        for (int i = 0; i < kBins; ++i) {
            float v = smarr[i];
            if (v > best) { best = v; bi = i; }
        }
        out[blockIdx.x] = -((2.0f * kPi) * (float)bi / (float)kBins - kPi);
    }
}

extern "C" void kernel_launch(void* const* d_in, const int* in_sizes, int n_in,
                              void* d_out, int out_size, void* d_ws, size_t ws_size,
                              hipStream_t stream) {
    (void)n_in; (void)d_ws; (void)ws_size; (void)out_size;
    const float* patches = (const float*)d_in[0];
    float* out = (float*)d_out;
    const int n_patches = in_sizes[0] / kNPix;  // 65536 patches of 32x32
    patch_dominant_orientation<<<dim3(n_patches), dim3(256), 0, stream>>>(patches, out);
}